// GAT_54443005444261
// MI455X (gfx1250) — compile-verified
//
#include <hip/hip_runtime.h>
#include <math.h>

#define NN 50000
#define DD 96
#define EE 800000
#define GG 1000
#define NHEADS 4
#define DHH 24
#define LAYERS 3
#define ETOT (EE + NN)
#define NEG_SLOPE 0.2f
#define BN_EPS 1e-5f

// LDS layouts for the WMMA GEMM
#define XL_STRIDE 108   // 96 DWORDs/row + 4 pad DWORDs every 32 -> conflict-free
#define WT_STRIDE 100   // transposed W, stride 100 -> conflict-free ds_load_b64

typedef __attribute__((ext_vector_type(2))) float v2f;
typedef __attribute__((ext_vector_type(8))) float v8f;
typedef __attribute__((ext_vector_type(4))) unsigned int u32x4;
typedef __attribute__((ext_vector_type(4))) int i32x4;
typedef __attribute__((ext_vector_type(8))) int i32x8;

// Generic pointers to LDS carry the wave-relative LDS byte offset in their low
// 32 bits (ISA 10.2 aperture mapping).
__device__ __forceinline__ unsigned lds_byte_addr(const void* p) {
    return (unsigned)(unsigned long long)p;
}

// ---------------------------------------------------------------------------
// float atomic max via order-preserving int trick (buffer init to -inf bits)
// ---------------------------------------------------------------------------
__device__ inline void atomicMaxF(float* addr, float val) {
    int vi = __float_as_int(val);
    if (vi >= 0) {
        atomicMax((int*)addr, vi);
    } else {
        atomicMin((unsigned int*)addr, (unsigned int)vi);
    }
}

// ---------------------------------------------------------------------------
// fill buffer with a raw u32 pattern (used for -inf init of segment-max)
// ---------------------------------------------------------------------------
__global__ void fill_u32(unsigned int* __restrict__ p, unsigned int v, int n) {
    int i = blockIdx.x * blockDim.x + threadIdx.x;
    if (i < n) p[i] = v;
}

// ---------------------------------------------------------------------------
// Y[nrows,96] = X[nrows,96] @ W[96,96] (+bias) via V_WMMA_F32_16X16X4_F32.
// Block = 128 threads = 4 wave32; each wave owns a 16x96 output slab
// (6 accumulator tiles of 16x16), K swept in 24 steps of 4.
//
// Data staging (CDNA5-specific):
//  * Wave 0 issues a TENSOR_LOAD_TO_LDS DMA of the block's 64x96 fp32 X tile.
//    TDM padding (pad_interval=32dw, pad_amount=4dw) yields an LDS row stride
//    of 108 DWORDs, so A-fragment ds_load_b64 reads are bank-conflict-free
//    (44*m mod 64 is a lane permutation; +2 upper-half offset cannot alias).
//    tensor_dim1 = remaining rows -> hardware zero-fill on the tail block.
//  * Concurrently, all 128 threads stage W transposed (Wt[col][k], stride 100)
//    so each B fragment is one aligned conflict-free ds_load_b64.
//  * Wave 0: s_wait_tensorcnt 0, then block barrier.
//
// Fragment layouts per ISA 7.12.2 (wave32):
//  A 16x4 f32: lanes 0-15 hold K=k0,k0+1; lanes 16-31 hold K=k0+2,k0+3; M=lane%16
//  B 4x16 f32: mirrored K split across lane halves, N striped over lanes
//  C/D 16x16 f32: VGPR r: lanes 0-15 -> M=r, lanes 16-31 -> M=r+8, N=lane%16
// ---------------------------------------------------------------------------
__global__ __launch_bounds__(128)
void gemm_n96_wmma(const float* __restrict__ X, const float* __restrict__ W,
                   const float* __restrict__ bias, float* __restrict__ Y, int nrows) {
    __shared__ float Xl[64 * XL_STRIDE];
    __shared__ float Wt[DD * WT_STRIDE];

    const int rowBase0 = blockIdx.x * 64;
    const int wave = threadIdx.x >> 5;
    const int lane = threadIdx.x & 31;

#if __has_builtin(__builtin_amdgcn_tensor_load_to_lds)
    if (wave == 0) {
        const int remi = nrows - rowBase0;
        const unsigned rem = (unsigned)(remi < 64 ? remi : 64);
        const unsigned long long ga =
            (unsigned long long)(X + (size_t)rowBase0 * DD);
        u32x4 g0;
        g0[0] = 1u;                                   // count=1 (valid D#)
        g0[1] = lds_byte_addr(Xl);                    // lds_addr
        g0[2] = (unsigned)(ga & 0xFFFFFFFFull);       // global_addr[31:0]
        g0[3] = (unsigned)((ga >> 32) & 0x1FFFFFFull) // global_addr[56:32]
                | (2u << 30);                         // type=2 ("image")
        i32x8 g1;
        g1[0] = (int)0x07120000u;  // data_size=4B, pad_en, interval=32dw, amount=4dw
        g1[1] = (int)(96u << 16);  // tensor_dim0 = 96
        g1[2] = (int)(rem << 16);  // tensor_dim1 = remaining rows (OOB -> zero)
        g1[3] = (int)(96u << 16);  // tile_dim0 = 96
        g1[4] = 64;                // tile_dim1 = 64
        g1[5] = 96;                // tensor_dim0_stride = 96
        g1[6] = 0;
        g1[7] = 0;
        i32x4 gz = {0, 0, 0, 0};
#if defined(__clang_major__) && __clang_major__ >= 23
        i32x8 gz8 = {0, 0, 0, 0, 0, 0, 0, 0};
        __builtin_amdgcn_tensor_load_to_lds(g0, g1, gz, gz, gz8, 0);
#else
        __builtin_amdgcn_tensor_load_to_lds(g0, g1, gz, gz, 0);
#endif
    }
#else
    // fallback: manual padded staging of the X tile
    for (int i = threadIdx.x; i < 64 * DD; i += 128) {
        int r = i / DD, k = i % DD;
        int row = rowBase0 + r;
        Xl[r * XL_STRIDE + k + ((k >> 5) << 2)] =
            (row < nrows) ? X[(size_t)row * DD + k] : 0.f;
    }
#endif

    // stage W transposed while the TDM DMA is in flight
    for (int i = threadIdx.x; i < DD * DD; i += 128) {
        int k = i / DD, col = i % DD;
        Wt[col * WT_STRIDE + k] = W[i];
    }

#if __has_builtin(__builtin_amdgcn_tensor_load_to_lds)
    if (wave == 0) {
#if __has_builtin(__builtin_amdgcn_s_wait_tensorcnt)
        __builtin_amdgcn_s_wait_tensorcnt(0);
#else
        asm volatile("s_wait_tensorcnt 0x0" ::: "memory");
#endif
    }
#endif
    __syncthreads();

    const int rowBase = rowBase0 + wave * 16;
    if (rowBase >= nrows) return;  // whole-wave uniform skip (nrows % 16 == 0)

    const int half = lane >> 4;  // 0: lanes 0-15, 1: lanes 16-31
    const int m    = lane & 15;

    const v8f vzero = {0.f, 0.f, 0.f, 0.f, 0.f, 0.f, 0.f, 0.f};
    v8f acc[6];
#pragma unroll
    for (int c = 0; c < 6; ++c) acc[c] = vzero;

    const float* xl = &Xl[(wave * 16 + m) * XL_STRIDE];
#pragma unroll 4
    for (int k0 = 0; k0 < DD; k0 += 4) {
        const int kk   = k0 + half * 2;
        const int kpad = kk + ((kk >> 5) << 2);      // padded LDS offset
        const v2f a = *(const v2f*)(xl + kpad);      // ds_load_b64, conflict-free
#pragma unroll
        for (int c = 0; c < 6; ++c) {
            const int col = c * 16 + m;
            const v2f b = *(const v2f*)&Wt[col * WT_STRIDE + kk];  // ds_load_b64
            acc[c] = __builtin_amdgcn_wmma_f32_16x16x4_f32(
                false, a, false, b, (short)0, acc[c], false, false);
        }
    }

#pragma unroll
    for (int c = 0; c < 6; ++c) {
        const int col = c * 16 + m;
        const float bb = bias ? bias[col] : 0.f;
#pragma unroll
        for (int r = 0; r < 8; ++r) {
            Y[(size_t)(rowBase + half * 8 + r) * DD + col] = acc[c][r] + bb;
        }
    }
}

// ---------------------------------------------------------------------------
// per-node attention logits: alpha_src/dst[n,h] = <h[n, h*24:(h+1)*24], att>
// ---------------------------------------------------------------------------
__global__ void alpha_kernel(const float* __restrict__ h,
                             const float* __restrict__ as,
                             const float* __restrict__ ad,
                             float* __restrict__ oas, float* __restrict__ oad) {
    int i = blockIdx.x * blockDim.x + threadIdx.x;  // node*NHEADS + head
    if (i >= NN * NHEADS) return;
    int node = i >> 2, head = i & 3;
    const float* hp  = h  + (size_t)node * DD + head * DHH;
    const float* asp = as + head * DHH;
    const float* adp = ad + head * DHH;
    float s0 = 0.f, s1 = 0.f;
#pragma unroll
    for (int j = 0; j < DHH; ++j) {
        float v = hp[j];
        s0 += v * asp[j];
        s1 += v * adp[j];
    }
    oas[i] = s0;
    oad[i] = s1;
}

// ---------------------------------------------------------------------------
// pass 1: e = leakyrelu(alpha_src[src]+alpha_dst[dst]); segment max into nmax
// ---------------------------------------------------------------------------
__global__ void edge_max_kernel(const int* __restrict__ ei,
                                const float* __restrict__ asrc,
                                const float* __restrict__ adst,
                                float* __restrict__ eed, float* __restrict__ nmax) {
    int e = blockIdx.x * blockDim.x + threadIdx.x;
    if (e >= ETOT) return;
    int s, d;
    if (e < EE) { s = ei[e]; d = ei[EE + e]; } else { s = d = e - EE; }
    const float4 a4 = *(const float4*)(asrc + (size_t)s * NHEADS);
    const float4 b4 = *(const float4*)(adst + (size_t)d * NHEADS);
    float ev[4] = {a4.x + b4.x, a4.y + b4.y, a4.z + b4.z, a4.w + b4.w};
    float* ep = eed + (size_t)e * NHEADS;
#pragma unroll
    for (int hh = 0; hh < NHEADS; ++hh) {
        float v = ev[hh] >= 0.f ? ev[hh] : NEG_SLOPE * ev[hh];
        ep[hh] = v;
        atomicMaxF(&nmax[d * NHEADS + hh], v);
    }
}

// ---------------------------------------------------------------------------
// pass 2: e = exp(e - max[dst]); segment sum into denom
// ---------------------------------------------------------------------------
__global__ void edge_exp_kernel(const int* __restrict__ ei, float* __restrict__ eed,
                                const float* __restrict__ nmax, float* __restrict__ den) {
    int e = blockIdx.x * blockDim.x + threadIdx.x;
    if (e >= ETOT) return;
    int d = (e < EE) ? ei[EE + e] : (e - EE);
    float* ep = eed + (size_t)e * NHEADS;
    const float* mp = nmax + (size_t)d * NHEADS;
#pragma unroll
    for (int hh = 0; hh < NHEADS; ++hh) {
        float v = __expf(ep[hh] - mp[hh]);
        ep[hh] = v;
        atomicAdd(&den[d * NHEADS + hh], v);
    }
}

// ---------------------------------------------------------------------------
// pass 3: agg[dst] += h[src] * (e / denom[dst]); one thread per (edge, 4-col chunk)
// ---------------------------------------------------------------------------
__global__ void edge_scatter_kernel(const int* __restrict__ ei,
                                    const float* __restrict__ h,
                                    const float* __restrict__ eed,
                                    const float* __restrict__ den,
                                    float* __restrict__ agg) {
    int t = blockIdx.x * blockDim.x + threadIdx.x;  // ETOT*24 < 2^31
    if (t >= ETOT * 24) return;
    int e = t / 24, q = t % 24;
    int s, d;
    if (e < EE) { s = ei[e]; d = ei[EE + e]; } else { s = d = e - EE; }
    int head = q / 6;  // 24 float4 chunks of 96 cols -> 6 chunks per head
    float alpha = eed[(size_t)e * NHEADS + head] /
                  (den[(size_t)d * NHEADS + head] + 1e-16f);
    const float4 hv = ((const float4*)(h + (size_t)s * DD))[q];
    float* ap = agg + (size_t)d * DD + q * 4;
    atomicAdd(ap + 0, hv.x * alpha);
    atomicAdd(ap + 1, hv.y * alpha);
    atomicAdd(ap + 2, hv.z * alpha);
    atomicAdd(ap + 3, hv.w * alpha);
}

// ---------------------------------------------------------------------------
// node-wise: out = relu(bn(agg + conv_bias))
// ---------------------------------------------------------------------------
__global__ void bn_relu_kernel(const float* __restrict__ agg, const float* __restrict__ bc,
                               const float* __restrict__ g, const float* __restrict__ be,
                               const float* __restrict__ mu, const float* __restrict__ var,
                               float* __restrict__ out) {
    int i = blockIdx.x * blockDim.x + threadIdx.x;
    if (i >= NN * DD) return;
    int c = i % DD;
    float v = agg[i] + bc[c];
    v = (v - mu[c]) * rsqrtf(var[c] + BN_EPS) * g[c] + be[c];
    out[i] = v > 0.f ? v : 0.f;
}

// ---------------------------------------------------------------------------
// global mean pool (sums + counts); one thread per (node, 4-col chunk)
// ---------------------------------------------------------------------------
__global__ void pool_kernel(const float* __restrict__ h, const int* __restrict__ batch,
                            float* __restrict__ pooled, float* __restrict__ counts) {
    int t = blockIdx.x * blockDim.x + threadIdx.x;
    if (t >= NN * 24) return;
    int node = t / 24, q = t % 24;
    int g = batch[node];
    const float4 hv = ((const float4*)(h + (size_t)node * DD))[q];
    float* pp = pooled + (size_t)g * DD + q * 4;
    atomicAdd(pp + 0, hv.x);
    atomicAdd(pp + 1, hv.y);
    atomicAdd(pp + 2, hv.z);
    atomicAdd(pp + 3, hv.w);
    if (q == 0) atomicAdd(&counts[g], 1.0f);
}

// ---------------------------------------------------------------------------
// per-graph MLP head: one block (128 threads) per graph; G=1000, trivial FLOPs
// ---------------------------------------------------------------------------
__global__ __launch_bounds__(128)
void mlp_kernel(const float* __restrict__ pooled, const float* __restrict__ counts,
                const float* __restrict__ W1, const float* __restrict__ b1,
                const float* __restrict__ W2, const float* __restrict__ b2,
                const float* __restrict__ W3, const float* __restrict__ b3,
                float* __restrict__ out) {
    __shared__ float p[DD];
    __shared__ float z1[DD];
    __shared__ float z2[DD / 2];
    int g = blockIdx.x, t = threadIdx.x;
    float cnt = fmaxf(counts[g], 1.0f);
    if (t < DD) p[t] = pooled[(size_t)g * DD + t] / cnt;
    __syncthreads();
    if (t < DD) {
        float s = b1[t];
        for (int k = 0; k < DD; ++k) s += p[k] * W1[k * DD + t];
        z1[t] = s > 0.f ? s : 0.f;
    }
    __syncthreads();
    if (t < DD / 2) {
        float s = b2[t];
        for (int k = 0; k < DD; ++k) s += z1[k] * W2[k * (DD / 2) + t];
        z2[t] = s > 0.f ? s : 0.f;
    }
    __syncthreads();
    if (t == 0) {
        float s = b3[0];
        for (int k = 0; k < DD / 2; ++k) s += z2[k] * W3[k];
        out[g] = s;
    }
}

// ---------------------------------------------------------------------------
extern "C" void kernel_launch(void* const* d_in, const int* in_sizes, int n_in,
                              void* d_out, int out_size, void* d_ws, size_t ws_size,
                              hipStream_t stream) {
    (void)in_sizes; (void)n_in; (void)out_size; (void)ws_size;

    const float* x     = (const float*)d_in[0];
    const int*   ei    = (const int*)d_in[1];
    const int*   batch = (const int*)d_in[2];
    const float* Wp    = (const float*)d_in[3];
    const float* bp    = (const float*)d_in[4];
    const float* Wc    = (const float*)d_in[5];
    const float* att_s = (const float*)d_in[6];
    const float* att_d = (const float*)d_in[7];
    const float* bc    = (const float*)d_in[8];
    const float* bng   = (const float*)d_in[9];
    const float* bnb   = (const float*)d_in[10];
    const float* bnm   = (const float*)d_in[11];
    const float* bnv   = (const float*)d_in[12];
    const float* W1    = (const float*)d_in[13];
    const float* b1    = (const float*)d_in[14];
    const float* W2    = (const float*)d_in[15];
    const float* b2    = (const float*)d_in[16];
    const float* W3    = (const float*)d_in[17];
    const float* b3    = (const float*)d_in[18];
    float* out = (float*)d_out;

    // carve workspace (256B aligned slices)
    char* ws = (char*)d_ws;
    size_t off = 0;
    auto carve = [&](size_t elems) -> float* {
        float* p = (float*)(ws + off);
        off += ((elems * sizeof(float) + 255) & ~(size_t)255);
        return p;
    };
    float* hA     = carve((size_t)NN * DD);      // layer input features
    float* hB     = carve((size_t)NN * DD);      // projected features h = x @ W
    float* hC     = carve((size_t)NN * DD);      // aggregated messages
    float* asrc   = carve((size_t)NN * NHEADS);
    float* adst   = carve((size_t)NN * NHEADS);
    float* nmax   = carve((size_t)NN * NHEADS);
    float* den    = carve((size_t)NN * NHEADS);
    float* eed    = carve((size_t)ETOT * NHEADS);
    float* pooled = carve((size_t)GG * DD);
    float* counts = carve((size_t)GG);

    const dim3 blk(256);
    const int gemmBlocks = (NN + 63) / 64;  // 782 blocks x 4 waves x 16 rows

    // input projection: hA = x @ Wp + bp
    hipLaunchKernelGGL(gemm_n96_wmma, dim3(gemmBlocks), dim3(128), 0, stream,
                       x, Wp, bp, hA, NN);

    for (int l = 0; l < LAYERS; ++l) {
        // hB = hA @ Wc[l]  (conv bias applied post-aggregation, per reference)
        hipLaunchKernelGGL(gemm_n96_wmma, dim3(gemmBlocks), dim3(128), 0, stream,
                           hA, Wc + (size_t)l * DD * DD, (const float*)nullptr, hB, NN);
        hipLaunchKernelGGL(alpha_kernel, dim3((NN * NHEADS + 255) / 256), blk, 0, stream,
                           hB, att_s + l * NHEADS * DHH, att_d + l * NHEADS * DHH,
                           asrc, adst);
        hipLaunchKernelGGL(fill_u32, dim3((NN * NHEADS + 255) / 256), blk, 0, stream,
                           (unsigned int*)nmax, 0xFF800000u, NN * NHEADS);  // -inf
        hipMemsetAsync(den, 0, (size_t)NN * NHEADS * sizeof(float), stream);
        hipMemsetAsync(hC, 0, (size_t)NN * DD * sizeof(float), stream);
        hipLaunchKernelGGL(edge_max_kernel, dim3((ETOT + 255) / 256), blk, 0, stream,
                           ei, asrc, adst, eed, nmax);
        hipLaunchKernelGGL(edge_exp_kernel, dim3((ETOT + 255) / 256), blk, 0, stream,
                           ei, eed, nmax, den);
        hipLaunchKernelGGL(edge_scatter_kernel, dim3((ETOT * 24 + 255) / 256), blk, 0,
                           stream, ei, hB, eed, den, hC);
        hipLaunchKernelGGL(bn_relu_kernel, dim3((NN * DD + 255) / 256), blk, 0, stream,
                           hC, bc + l * DD, bng + l * DD, bnb + l * DD,
                           bnm + l * DD, bnv + l * DD, hA);
    }

    hipMemsetAsync(pooled, 0, (size_t)GG * DD * sizeof(float), stream);
    hipMemsetAsync(counts, 0, (size_t)GG * sizeof(float), stream);
    hipLaunchKernelGGL(pool_kernel, dim3((NN * 24 + 255) / 256), blk, 0, stream,
                       hA, batch, pooled, counts);
    hipLaunchKernelGGL(mlp_kernel, dim3(GG), dim3(128), 0, stream,
                       pooled, counts, W1, b1, W2, b2, W3, b3, out);
}